// ImpulseNoise_12275016532591
// MI455X (gfx1250) — compile-verified
//
#include <hip/hip_runtime.h>

// Salt & pepper (impulse) noise, MI455X / gfx1250.
// Pass 1: streaming clamp-copy x -> out. NT B128 loads (don't pollute L2),
//         RT B128 stores (output stays resident in the 192MB L2).
//         Fast path issues 4 loads as a clause before any store so the wave
//         keeps 4 LOADcnt entries in flight instead of load->wait->store.
// Pass 2: per-batch scatter of the salt/pepper overwrites; random 4B stores
//         mostly hit L2 because pass 1 left the output lines resident.

typedef __attribute__((ext_vector_type(4))) float v4f;

__device__ __forceinline__ v4f clamp01(v4f v) {
  v.x = fminf(fmaxf(v.x, 0.0f), 1.0f);   // folds to v_max_num_f32 ... clamp
  v.y = fminf(fmaxf(v.y, 0.0f), 1.0f);
  v.z = fminf(fmaxf(v.z, 0.0f), 1.0f);
  v.w = fminf(fmaxf(v.w, 0.0f), 1.0f);
  return v;
}

// ---------------------------------------------------------------------------
// Pass 1: out[i] = clamp(x[i], 0, 1), 4x float4 per lane.
// ---------------------------------------------------------------------------
__global__ __launch_bounds__(256) void impulse_copy_clamp(
    const float* __restrict__ x, float* __restrict__ out, long long n4) {
  const long long bd   = blockDim.x;
  const long long tile = bd * 4;
  const long long base = (long long)blockIdx.x * tile + threadIdx.x;

  const v4f* __restrict__ xv = (const v4f*)x;
  v4f* __restrict__ ov = (v4f*)out;

  if (base + 3 * bd < n4) {
    // Interior tile (all blocks for the reference shape): issue all 4 NT B128
    // loads back-to-back so they form a clause with staggered loadcnt waits.
    v4f a = __builtin_nontemporal_load(&xv[base]);
    v4f b = __builtin_nontemporal_load(&xv[base + bd]);
    v4f c = __builtin_nontemporal_load(&xv[base + 2 * bd]);
    v4f d = __builtin_nontemporal_load(&xv[base + 3 * bd]);
    ov[base]          = clamp01(a);
    ov[base + bd]     = clamp01(b);
    ov[base + 2 * bd] = clamp01(c);
    ov[base + 3 * bd] = clamp01(d);
  } else {
    // Tail fallback (unused for the reference shape, kept for generality).
#pragma unroll
    for (int k = 0; k < 4; ++k) {
      long long i = base + (long long)k * bd;
      if (i < n4) {
        v4f v = __builtin_nontemporal_load(&xv[i]);
        ov[i] = clamp01(v);
      }
    }
  }
}

// ---------------------------------------------------------------------------
// Pass 2: for batch b (blockIdx.y), position j in [0, num_sp):
//   out[b*per_img + indices[b, j]] = (j < num_salt[b]) ? 1.0f : 0.0f
// indices are a permutation slice -> no duplicates -> order-independent.
// num_salt[b] is wave-uniform (b = blockIdx.y) -> scalar load.
// ---------------------------------------------------------------------------
__global__ __launch_bounds__(256) void impulse_scatter(
    const int* __restrict__ indices, const int* __restrict__ num_salt,
    float* __restrict__ out, int num_sp, long long per_img) {
  const int b = blockIdx.y;
  const int j = blockIdx.x * blockDim.x + threadIdx.x;
  if (j < num_sp) {
    const int ns = num_salt[b];                        // uniform -> SGPR
    const int px = indices[(long long)b * num_sp + j]; // coalesced B32 load
    const float v = (j < ns) ? 1.0f : 0.0f;
    out[(long long)b * per_img + (long long)px] = v;   // scattered B32 store
  }
}

extern "C" void kernel_launch(void* const* d_in, const int* in_sizes, int n_in,
                              void* d_out, int out_size, void* d_ws, size_t ws_size,
                              hipStream_t stream) {
  (void)n_in; (void)d_ws; (void)ws_size; (void)out_size;

  const float* x        = (const float*)d_in[0];  // [B, C, H, W] f32
  const int*   indices  = (const int*)d_in[1];    // [B, NUM_SP] i32
  const int*   num_salt = (const int*)d_in[2];    // [B] i32
  float*       out      = (float*)d_out;          // [B, C, H, W] f32

  const long long n_elems = (long long)in_sizes[0];       // B*C*H*W
  const int       B       = in_sizes[2];                  // 64
  const int       num_sp  = in_sizes[1] / B;              // 55050
  const long long per_img = n_elems / B;                  // C*H*W
  const long long n4      = n_elems / 4;                  // float4 count (exact)

  // Pass 1: clamp-copy. 256 threads * 4 float4 each = 4096 floats per block.
  {
    const int  threads = 256;
    const long long per_block = (long long)threads * 4;
    const int  blocks = (int)((n4 + per_block - 1) / per_block);
    impulse_copy_clamp<<<blocks, threads, 0, stream>>>(x, out, n4);
  }

  // Pass 2: scatter overwrites (serialized after pass 1 on the same stream).
  {
    const int threads = 256;
    dim3 grid((num_sp + threads - 1) / threads, B, 1);
    impulse_scatter<<<grid, threads, 0, stream>>>(indices, num_salt, out,
                                                  num_sp, per_img);
  }
}